// Temporal_Attention_layer_3539053051906
// MI455X (gfx1250) — compile-verified
//
#include <hip/hip_runtime.h>
#include <hip/hip_bf16.h>

// Temporal attention layer for MI455X (gfx1250, wave32).
//
// Roofline: x = (32,2048,64,64) fp32 = 1 GiB dominates (≈46us @ 23.3 TB/s).
// Total math ≈ 2.1 GFLOP -> compute is trivial vs WMMA ceiling. Strategy:
//   pass1: stream x exactly once, produce both contractions
//          acc1[b,c,t] = sum_v U1[v] x[b,v,c,t]   (512 KB, atomically reduced)
//          rhs[b,v,t]  = sum_c U3[c] x[b,v,c,t]   (16 MB, lives in L2)
//   pass2: per-b fused WMMA (V_WMMA_F32_16X16X4_F32) pipeline:
//          for each 64-wide v chunk: L = acc1^T @ U2chunk ; P += L @ Rchunk
//          then S = sigmoid(P + b_e); E = V_e @ S; softmax over axis=1 (t).

typedef __attribute__((ext_vector_type(2))) float v2f;
typedef __attribute__((ext_vector_type(8))) float v8f;

namespace {
constexpr int kB  = 32;
constexpr int kV  = 2048;
constexpr int kC  = 64;
constexpr int kT  = 64;
constexpr int kVC = 64;   // v-values per chunk / per pass1 block
constexpr int XS  = 68;   // padded LDS row stride (floats); keeps float4 16B-aligned
}

__device__ __forceinline__ v8f wmma4_f32(v2f a, v2f b, v8f c) {
  // D = A(16x4) * B(4x16) + C(16x16), fp32, wave32
  return __builtin_amdgcn_wmma_f32_16x16x4_f32(false, a, false, b, (short)0, c,
                                               false, false);
}

__global__ void ta_zero(float* __restrict__ p, int n) {
  int i = blockIdx.x * blockDim.x + threadIdx.x;
  if (i < n) p[i] = 0.0f;
}

// ---------------------------------------------------------------------------
// Pass 1: stream x once. Grid = (b, vchunk) = (32, 32), 256 threads.
// Each thread owns 16 fixed (c,t) slots of the acc1 tile in registers.
// ---------------------------------------------------------------------------
__global__ __launch_bounds__(256) void ta_pass1(
    const float* __restrict__ x, const float* __restrict__ U1,
    const float* __restrict__ U3, float* __restrict__ acc1,
    float* __restrict__ rhs) {
  __shared__ float X[kC * XS];  // tile x[b,v,:,:] as [c][t]

  const int tid = threadIdx.x;
  const int b   = blockIdx.x;
  const int v0  = blockIdx.y * kVC;

  // float4 index mapping: idx = tid + j*256 (j=0..3) -> c = idx>>4, t4 = (idx&15)*4
  float4 pacc[4] = {};

  const float* xb = x + (size_t)(b * kV + v0) * (kC * kT);

  for (int v = 0; v < kVC; ++v) {
    const float4* src = (const float4*)(xb + (size_t)v * (kC * kT));
    float4 xr[4];
#pragma unroll
    for (int j = 0; j < 4; ++j) xr[j] = src[tid + j * 256];

    // stash tile in LDS for the rhs column reduction
#pragma unroll
    for (int j = 0; j < 4; ++j) {
      int idx = tid + j * 256;
      int c = idx >> 4, t4 = (idx & 15) * 4;
      *(float4*)&X[c * XS + t4] = xr[j];
    }

    // acc1 partial: register accumulate (U1[v] is wave-uniform -> SGPR)
    const float u1 = U1[v0 + v];
#pragma unroll
    for (int j = 0; j < 4; ++j) {
      pacc[j].x = __builtin_fmaf(u1, xr[j].x, pacc[j].x);
      pacc[j].y = __builtin_fmaf(u1, xr[j].y, pacc[j].y);
      pacc[j].z = __builtin_fmaf(u1, xr[j].z, pacc[j].z);
      pacc[j].w = __builtin_fmaf(u1, xr[j].w, pacc[j].w);
    }

    __syncthreads();
    if (tid < kT) {  // rhs[b, v0+v, t] = sum_c U3[c] * x[c][t]
      float s = 0.0f;
#pragma unroll
      for (int c = 0; c < kC; ++c) s = __builtin_fmaf(U3[c], X[c * XS + tid], s);
      rhs[(size_t)(b * kV + v0 + v) * kT + tid] = s;  // coalesced 256B store
    }
    __syncthreads();
  }

  // flush acc1 partial tile (32 v-chunks contend per location -> cheap)
#pragma unroll
  for (int j = 0; j < 4; ++j) {
    int idx = tid + j * 256;
    int c = idx >> 4, t = (idx & 15) * 4;
    float* dst = acc1 + (size_t)b * (kC * kT) + c * kT + t;
    atomicAdd(dst + 0, pacc[j].x);
    atomicAdd(dst + 1, pacc[j].y);
    atomicAdd(dst + 2, pacc[j].z);
    atomicAdd(dst + 3, pacc[j].w);
  }
}

// ---------------------------------------------------------------------------
// Pass 2: one block per b, 512 threads = 16 wave32s, one 16x16 D-tile per wave.
// All three GEMMs use V_WMMA_F32_16X16X4_F32.
// ---------------------------------------------------------------------------
__global__ __launch_bounds__(512) void ta_pass2(
    const float* __restrict__ U2, const float* __restrict__ be,
    const float* __restrict__ Ve, const float* __restrict__ acc1,
    const float* __restrict__ rhs, float* __restrict__ out) {
  __shared__ float Atc[kC * XS];  // acc1[b] as [c][t]  (A^T view: A[t][c]=Atc[c][t])
  __shared__ float Ul[kC * XS];   // U2 chunk [c][vc]; later V_e[b] [t][s]
  __shared__ float Rl[kVC * XS];  // rhs chunk [vc][t]; later E [t][u]
  __shared__ float Ll[kT * XS];   // lhs chunk [t][vc]; later S=sigmoid(P+b_e) [t][u]

  const int tid  = threadIdx.x;
  const int lane = tid & 31;
  const int wave = tid >> 5;
  const int b    = blockIdx.x;

  // wave32 WMMA fragment geometry (CDNA5 ISA 7.12.2, 32-bit layouts)
  const int l15 = lane & 15;
  const int kh  = (lane >> 4) * 2;  // K-pair offset per lane half (A and B frags)
  const int ch  = (lane >> 4) * 8;  // C/D row offset per lane half

  // one 16x16 tile per wave in the 4x4 tile grid of the 64x64 outputs
  const int m0 = (wave >> 2) * 16;
  const int n0 = (wave & 3) * 16;

  // load acc1[b] (4096 floats) into LDS
  {
    const float4* src = (const float4*)(acc1 + (size_t)b * (kC * kT));
#pragma unroll
    for (int j = 0; j < 2; ++j) {
      int idx = tid + j * 512;  // float4 index 0..1023
      int c = idx >> 4, t4 = (idx & 15) * 4;
      *(float4*)&Atc[c * XS + t4] = src[idx];
    }
  }

  v8f prod = {};  // product[b] tile accumulator, persists across v chunks

  for (int chunk = 0; chunk < kV / kVC; ++chunk) {
    // cooperative loads: U2[:, chunk] and rhs[b, chunk, :]
    {
      const float* u2base = U2 + chunk * kVC;  // row stride kV
#pragma unroll
      for (int j = 0; j < 2; ++j) {
        int idx = tid + j * 512;
        int c = idx >> 4, v4 = (idx & 15) * 4;
        *(float4*)&Ul[c * XS + v4] = *(const float4*)(u2base + c * kV + v4);
      }
      const float4* rsrc =
          (const float4*)(rhs + (size_t)(b * kV + chunk * kVC) * kT);
#pragma unroll
      for (int j = 0; j < 2; ++j) {
        int idx = tid + j * 512;
        int vc = idx >> 4, t4 = (idx & 15) * 4;
        *(float4*)&Rl[vc * XS + t4] = rsrc[idx];
      }
    }
    __syncthreads();

    // phase 1: L(t,vc) = sum_c A(t,c) * U2(c,vc), A(t,c) = Atc[c][t]
    v8f lacc = {};
#pragma unroll
    for (int k = 0; k < kC; k += 4) {
      v2f a, bb;
      a.x  = Atc[(k + kh + 0) * XS + (m0 + l15)];
      a.y  = Atc[(k + kh + 1) * XS + (m0 + l15)];
      bb.x = Ul[(k + kh + 0) * XS + (n0 + l15)];
      bb.y = Ul[(k + kh + 1) * XS + (n0 + l15)];
      lacc = wmma4_f32(a, bb, lacc);
    }
    // write this wave's L tile to LDS (D layout: vgpr i <-> rows i, i+8)
#pragma unroll
    for (int i = 0; i < 8; ++i) Ll[(m0 + ch + i) * XS + (n0 + l15)] = lacc[i];
    __syncthreads();

    // phase 2: P(t,u) += sum_vc L(t,vc) * R(vc,u)
#pragma unroll
    for (int k = 0; k < kVC; k += 4) {
      v2f a, bb;
      a.x  = Ll[(m0 + l15) * XS + (k + kh + 0)];
      a.y  = Ll[(m0 + l15) * XS + (k + kh + 1)];
      bb.x = Rl[(k + kh + 0) * XS + (n0 + l15)];
      bb.y = Rl[(k + kh + 1) * XS + (n0 + l15)];
      prod = wmma4_f32(a, bb, prod);
    }
    __syncthreads();  // protect Ul/Rl/Ll before next chunk overwrites them
  }

  // ---- epilogue ----
  // product tiles -> Ll ; V_e[b] -> Ul
#pragma unroll
  for (int i = 0; i < 8; ++i) Ll[(m0 + ch + i) * XS + (n0 + l15)] = prod[i];
  {
    const float4* vsrc = (const float4*)(Ve + (size_t)b * (kT * kT));
#pragma unroll
    for (int j = 0; j < 2; ++j) {
      int idx = tid + j * 512;
      int t = idx >> 4, s4 = (idx & 15) * 4;
      *(float4*)&Ul[t * XS + s4] = vsrc[idx];
    }
  }
  __syncthreads();

  // S = sigmoid(P + b_e), in place in Ll
#pragma unroll
  for (int j = 0; j < 8; ++j) {
    int idx = tid + j * 512;  // 0..4095
    int t = idx >> 6, u = idx & 63;
    float p = Ll[t * XS + u] + be[idx];
    Ll[t * XS + u] = 1.0f / (1.0f + __expf(-p));
  }
  __syncthreads();

  // E(t,u) = sum_s Ve(t,s) * S(s,u)
  v8f eacc = {};
#pragma unroll
  for (int k = 0; k < kT; k += 4) {
    v2f a, bb;
    a.x  = Ul[(m0 + l15) * XS + (k + kh + 0)];
    a.y  = Ul[(m0 + l15) * XS + (k + kh + 1)];
    bb.x = Ll[(k + kh + 0) * XS + (n0 + l15)];
    bb.y = Ll[(k + kh + 1) * XS + (n0 + l15)];
    eacc = wmma4_f32(a, bb, eacc);
  }
#pragma unroll
  for (int i = 0; i < 8; ++i) Rl[(m0 + ch + i) * XS + (n0 + l15)] = eacc[i];
  __syncthreads();

  // softmax over axis=1 (t) per column u; coalesced stores (64 lanes per row t)
  if (tid < kT) {
    const int u = tid;
    float mx = -3.402823466e38f;
    for (int t = 0; t < kT; ++t) mx = fmaxf(mx, Rl[t * XS + u]);
    float s = 0.0f;
    for (int t = 0; t < kT; ++t) {
      float e = __expf(Rl[t * XS + u] - mx);
      Rl[t * XS + u] = e;
      s += e;
    }
    const float inv = 1.0f / s;
    for (int t = 0; t < kT; ++t)
      out[(size_t)b * (kT * kT) + t * kT + u] = Rl[t * XS + u] * inv;
  }
}

extern "C" void kernel_launch(void* const* d_in, const int* in_sizes, int n_in,
                              void* d_out, int out_size, void* d_ws,
                              size_t ws_size, hipStream_t stream) {
  const float* x  = (const float*)d_in[0];  // (B,V,C,T)
  const float* U1 = (const float*)d_in[1];  // (V)
  const float* U2 = (const float*)d_in[2];  // (C,V)
  const float* U3 = (const float*)d_in[3];  // (C)
  const float* be = (const float*)d_in[4];  // (1,T,T)
  const float* Ve = (const float*)d_in[5];  // (B,T,T)
  float* out = (float*)d_out;               // (B,T,T)

  // workspace layout: acc1 (512 KB) | rhs (16 MB)
  float* acc1 = (float*)d_ws;
  float* rhs  = acc1 + (size_t)kB * kC * kT;

  const int accN = kB * kC * kT;  // 131072
  ta_zero<<<(accN + 255) / 256, 256, 0, stream>>>(acc1, accN);

  dim3 g1(kB, kV / kVC);  // (32, 32)
  ta_pass1<<<g1, 256, 0, stream>>>(x, U1, U3, acc1, rhs);

  ta_pass2<<<kB, 512, 0, stream>>>(U2, be, Ve, acc1, rhs, out);
}